// Nodes2Adj_46273977647529
// MI455X (gfx1250) — compile-verified
//
#include <hip/hip_runtime.h>

// Problem constants (fixed by setup_inputs: B=4, H=W=128, n=32, P=16)
#define BB 4
#define NN 32
#define HH 128
#define WW 128
#define PP 16
#define RR (BB * NN * NN)   // 4096 RoIs

typedef float v2f __attribute__((ext_vector_type(2)));
typedef float v8f __attribute__((ext_vector_type(8)));

// ---------------------------------------------------------------------------
// K0: zero the BN stat accumulators (96 doubles)
// ---------------------------------------------------------------------------
__global__ void k_zero_stats(double* p) {
    int t = threadIdx.x;
    if (t < 96) p[t] = 0.0;
}

// ---------------------------------------------------------------------------
// K1: RoI max-pool + slope flip + focus band mask -> f [R,16,16]
// One block per RoI, one thread per bin.
// ---------------------------------------------------------------------------
__global__ void k_roipool(const float* __restrict__ img,
                          const float* __restrict__ nodes,
                          float* __restrict__ f) {
    int r   = blockIdx.x;
    int b   = r >> 10;           // / (n*n)
    int rem = r & 1023;
    int i   = rem >> 5;
    int j   = rem & 31;

    float ax = nodes[(b * NN + i) * 2 + 0], ay = nodes[(b * NN + i) * 2 + 1];
    float cx = nodes[(b * NN + j) * 2 + 0], cy = nodes[(b * NN + j) * 2 + 1];

    // rois = sorted coords scaled by [W,H,W,H]; jnp.round == rint (nearest-even)
    int x1 = (int)rintf(fminf(ax, cx) * (float)WW);
    int x2 = (int)rintf(fmaxf(ax, cx) * (float)WW);
    int y1 = (int)rintf(fminf(ay, cy) * (float)HH);
    int y2 = (int)rintf(fmaxf(ay, cy) * (float)HH);

    float slope = (cy - ay) / (cx - ax + 1e-6f);
    int right = (slope > 0.0f) ? 1 : 0;

    float bw = (float)max(x2 - x1 + 1, 1) / (float)PP;
    float bh = (float)max(y2 - y1 + 1, 1) / (float)PP;

    __shared__ float pooled[PP][PP];
    int t  = threadIdx.x;
    int py = t >> 4, px = t & 15;

    int wst = min(max((int)floorf((float)px * bw) + x1, 0), WW);
    int wen = min(max((int)ceilf((float)(px + 1) * bw) + x1, 0), WW);
    int hst = min(max((int)floorf((float)py * bh) + y1, 0), HH);
    int hen = min(max((int)ceilf((float)(py + 1) * bh) + y1, 0), HH);

    float v = 0.0f;                      // empty bin -> 0
    if (hen > hst && wen > wst) {
        const float* ib = img + (size_t)b * HH * WW;
        float m = -1e30f;
        for (int y = hst; y < hen; ++y)
            for (int x = wst; x < wen; ++x)
                m = fmaxf(m, ib[y * WW + x]);
        v = m;
    }
    pooled[py][px] = v;
    __syncthreads();

    int sx = right ? px : (PP - 1 - px);  // flip along width if slope <= 0
    float o = pooled[py][sx];
    int d = py - px; if (d < 0) d = -d;   // focus band |h - w| <= 4
    if (d > 4) o = 0.0f;
    f[(size_t)r * 256 + py * 16 + px] = o;
}

// ---------------------------------------------------------------------------
// K2: conv1 (1->32, 3x3 SAME) statistics pass: accumulate per-channel
// sum & sumsq (double atomics). One block per sample.
// Thread t: co = t>>3, strided over pixels.
// ---------------------------------------------------------------------------
__global__ void k_conv1_stats(const float* __restrict__ f,
                              const float* __restrict__ w,
                              const float* __restrict__ bias,
                              double* __restrict__ sum1,
                              double* __restrict__ sq1) {
    __shared__ float fs[PP][PP];
    __shared__ float rs[256], rq[256];
    int r = blockIdx.x, t = threadIdx.x;
    fs[t >> 4][t & 15] = f[(size_t)r * 256 + t];
    __syncthreads();

    int co = t >> 3, idx = t & 7;
    float w9[9];
#pragma unroll
    for (int q = 0; q < 9; ++q) w9[q] = w[co * 9 + q];
    float bco = bias[co];

    float ls = 0.0f, lq = 0.0f;
    for (int pp = idx; pp < 256; pp += 8) {
        int py = pp >> 4, px = pp & 15;
        float acc = bco;
#pragma unroll
        for (int dy = -1; dy <= 1; ++dy) {
            int yy = py + dy;
            if ((unsigned)yy < (unsigned)PP) {
#pragma unroll
                for (int dx = -1; dx <= 1; ++dx) {
                    int xx = px + dx;
                    if ((unsigned)xx < (unsigned)PP)
                        acc += w9[(dy + 1) * 3 + dx + 1] * fs[yy][xx];
                }
            }
        }
        ls += acc; lq += acc * acc;
    }
    rs[t] = ls; rq[t] = lq;
    __syncthreads();
    for (int off = 4; off >= 1; off >>= 1) {
        if (idx < off) { rs[t] += rs[t + off]; rq[t] += rq[t + off]; }
        __syncthreads();
    }
    if (idx == 0) {
        atomicAdd(&sum1[co], (double)rs[t]);
        atomicAdd(&sq1[co],  (double)rq[t]);
    }
}

// ---------------------------------------------------------------------------
// K3/K6: BN finalize -> per-channel scale/shift
// ---------------------------------------------------------------------------
__global__ void k_bn_finalize(const double* __restrict__ sum,
                              const double* __restrict__ sq,
                              const float* __restrict__ g,
                              const float* __restrict__ b,
                              float* __restrict__ scale,
                              float* __restrict__ shift,
                              int C, double count) {
    int c = threadIdx.x;
    if (c < C) {
        double mean = sum[c] / count;
        double var  = sq[c] / count - mean * mean;     // biased var (jnp default)
        double sc   = (double)g[c] / sqrt(var + 1e-5);
        scale[c] = (float)sc;
        shift[c] = (float)((double)b[c] - mean * sc);
    }
}

// ---------------------------------------------------------------------------
// K4: conv1 (recompute) + BN affine + ReLU + 2x2 maxpool -> h1 [R,32,8,8]
// ---------------------------------------------------------------------------
__global__ void k_conv1_apply(const float* __restrict__ f,
                              const float* __restrict__ w,
                              const float* __restrict__ bias,
                              const float* __restrict__ scale,
                              const float* __restrict__ shift,
                              float* __restrict__ h1) {
    __shared__ float fs[PP][PP];
    int r = blockIdx.x, t = threadIdx.x;
    fs[t >> 4][t & 15] = f[(size_t)r * 256 + t];
    __syncthreads();

    int co = t >> 3, idx = t & 7;
    float w9[9];
#pragma unroll
    for (int q = 0; q < 9; ++q) w9[q] = w[co * 9 + q];
    float bco = bias[co], sc = scale[co], sh = shift[co];

    for (int pos = idx; pos < 64; pos += 8) {
        int qy = pos >> 3, qx = pos & 7;
        float m = -1e30f;
#pragma unroll
        for (int sy = 0; sy < 2; ++sy) {
#pragma unroll
            for (int sxi = 0; sxi < 2; ++sxi) {
                int py = 2 * qy + sy, px = 2 * qx + sxi;
                float acc = bco;
#pragma unroll
                for (int dy = -1; dy <= 1; ++dy) {
                    int yy = py + dy;
                    if ((unsigned)yy < (unsigned)PP) {
#pragma unroll
                        for (int dx = -1; dx <= 1; ++dx) {
                            int xx = px + dx;
                            if ((unsigned)xx < (unsigned)PP)
                                acc += w9[(dy + 1) * 3 + dx + 1] * fs[yy][xx];
                        }
                    }
                }
                m = fmaxf(m, fmaxf(acc * sc + sh, 0.0f));
            }
        }
        h1[(size_t)r * 2048 + co * 64 + pos] = m;
    }
}

// ---------------------------------------------------------------------------
// K5: conv2 (32->16, 3x3 SAME on 8x8) statistics pass
// ---------------------------------------------------------------------------
__device__ __forceinline__ float conv2_at(const float* hs, const float* wsh,
                                          int co, int py, int px, float bco) {
    float acc = bco;
    for (int ci = 0; ci < 32; ++ci) {
        const float* fc = &hs[ci * 64];
        const float* wc = &wsh[co * 288 + ci * 9];
#pragma unroll
        for (int dy = -1; dy <= 1; ++dy) {
            int yy = py + dy;
            if ((unsigned)yy < 8u) {
#pragma unroll
                for (int dx = -1; dx <= 1; ++dx) {
                    int xx = px + dx;
                    if ((unsigned)xx < 8u)
                        acc += wc[(dy + 1) * 3 + dx + 1] * fc[yy * 8 + xx];
                }
            }
        }
    }
    return acc;
}

__global__ void k_conv2_stats(const float* __restrict__ h1,
                              const float* __restrict__ w,
                              const float* __restrict__ bias,
                              double* __restrict__ sum2,
                              double* __restrict__ sq2) {
    __shared__ float hs[32 * 64];     // 8 KB
    __shared__ float wsh[16 * 288];   // 18 KB
    __shared__ float rs[256], rq[256];
    int r = blockIdx.x, t = threadIdx.x;
    for (int q = t; q < 2048; q += 256) hs[q] = h1[(size_t)r * 2048 + q];
    for (int q = t; q < 4608; q += 256) wsh[q] = w[q];
    __syncthreads();

    int co = t >> 4, idx = t & 15;
    float bco = bias[co];
    float ls = 0.0f, lq = 0.0f;
    for (int pp = idx; pp < 64; pp += 16) {
        float acc = conv2_at(hs, wsh, co, pp >> 3, pp & 7, bco);
        ls += acc; lq += acc * acc;
    }
    rs[t] = ls; rq[t] = lq;
    __syncthreads();
    for (int off = 8; off >= 1; off >>= 1) {
        if (idx < off) { rs[t] += rs[t + off]; rq[t] += rq[t + off]; }
        __syncthreads();
    }
    if (idx == 0) {
        atomicAdd(&sum2[co], (double)rs[t]);
        atomicAdd(&sq2[co],  (double)rq[t]);
    }
}

// ---------------------------------------------------------------------------
// K7: conv2 (recompute) + BN affine + ReLU + 2x2 maxpool -> h2 [R,256]
// One thread per (co, pooled pos).
// ---------------------------------------------------------------------------
__global__ void k_conv2_apply(const float* __restrict__ h1,
                              const float* __restrict__ w,
                              const float* __restrict__ bias,
                              const float* __restrict__ scale,
                              const float* __restrict__ shift,
                              float* __restrict__ h2) {
    __shared__ float hs[32 * 64];
    __shared__ float wsh[16 * 288];
    int r = blockIdx.x, t = threadIdx.x;
    for (int q = t; q < 2048; q += 256) hs[q] = h1[(size_t)r * 2048 + q];
    for (int q = t; q < 4608; q += 256) wsh[q] = w[q];
    __syncthreads();

    int co = t >> 4, pos = t & 15;
    int qy = pos >> 2, qx = pos & 3;
    float bco = bias[co], sc = scale[co], sh = shift[co];
    float m = -1e30f;
#pragma unroll
    for (int sy = 0; sy < 2; ++sy) {
#pragma unroll
        for (int sxi = 0; sxi < 2; ++sxi) {
            float acc = conv2_at(hs, wsh, co, 2 * qy + sy, 2 * qx + sxi, bco);
            m = fmaxf(m, fmaxf(acc * sc + sh, 0.0f));
        }
    }
    h2[(size_t)r * 256 + co * 16 + pos] = m;   // flatten order = [16,4,4]
}

// ---------------------------------------------------------------------------
// K8: fused FC1 (WMMA f32 16x16x4) + bias + ReLU + FC2 + bias + clip.
// One wave (32 threads) per 16-row M-stripe. N = 64 -> 4 tiles of 16.
// A: h2 [M=4096, K=256] row-major.  B[k][n] = fc1_w[n][k] (fc1_w is [64,256]).
//
// f32 WMMA VGPR layouts (ISA 7.12.2):
//   A 16x4:  lane l -> M = l%16 ; VGPRv -> K = 2*(l/16) + v
//   B 4x16:  lane l -> N = l%16 ; VGPRv -> K = 2*(l/16) + v
//   C/D:     lane l -> N = l%16 ; VGPRv -> M = v + 8*(l/16)
// ---------------------------------------------------------------------------
__global__ void __launch_bounds__(32) k_fc_wmma(const float* __restrict__ h2,
                                                const float* __restrict__ w1,
                                                const float* __restrict__ b1,
                                                const float* __restrict__ w2,
                                                const float* __restrict__ b2,
                                                float* __restrict__ out) {
    int m0 = blockIdx.x * 16;
    int l  = threadIdx.x;
    int lm = l & 15;
    int hk = (l >> 4) * 2;

    v8f acc[4] = {};
    const float* arow = h2 + (size_t)(m0 + lm) * 256 + hk;

    for (int kk = 0; kk < 256; kk += 4) {
        v2f a;
        a.x = arow[kk];
        a.y = arow[kk + 1];
#pragma unroll
        for (int nt = 0; nt < 4; ++nt) {
            const float* brow = w1 + (size_t)(nt * 16 + lm) * 256 + kk + hk;
            v2f bf;
            bf.x = brow[0];
            bf.y = brow[1];
            acc[nt] = __builtin_amdgcn_wmma_f32_16x16x4_f32(
                false, a, false, bf, (short)0, acc[nt], false, false);
        }
    }

    __shared__ float hsh[16][64];
#pragma unroll
    for (int nt = 0; nt < 4; ++nt) {
        int n = nt * 16 + lm;
        float bn = b1[n];
#pragma unroll
        for (int v = 0; v < 8; ++v) {
            int m = v + (l >> 4) * 8;
            hsh[m][n] = fmaxf(acc[nt][v] + bn, 0.0f);   // ReLU(FC1)
        }
    }
    __syncthreads();

    if (l < 16) {
        float s = b2[0];
        for (int n = 0; n < 64; ++n) s += hsh[l][n] * w2[n];
        out[m0 + l] = fminf(fmaxf(s, 0.0f), 1.0f);       // FC2 + clip
    }
}

// ---------------------------------------------------------------------------
// Host launcher
// ---------------------------------------------------------------------------
extern "C" void kernel_launch(void* const* d_in, const int* in_sizes, int n_in,
                              void* d_out, int out_size, void* d_ws, size_t ws_size,
                              hipStream_t stream) {
    const float* img   = (const float*)d_in[0];
    const float* nodes = (const float*)d_in[1];
    const float* c1w   = (const float*)d_in[2];
    const float* c1b   = (const float*)d_in[3];
    const float* g1    = (const float*)d_in[4];
    const float* bb1   = (const float*)d_in[5];
    const float* c2w   = (const float*)d_in[6];
    const float* c2b   = (const float*)d_in[7];
    const float* g2    = (const float*)d_in[8];
    const float* bb2   = (const float*)d_in[9];
    const float* fc1w  = (const float*)d_in[10];
    const float* fc1b  = (const float*)d_in[11];
    const float* fc2w  = (const float*)d_in[12];
    const float* fc2b  = (const float*)d_in[13];
    float* out = (float*)d_out;

    // Workspace layout: 96 doubles (stats), then floats.
    double* dsum1 = (double*)d_ws;     // 32
    double* dsq1  = dsum1 + 32;        // 32
    double* dsum2 = dsq1 + 32;         // 16
    double* dsq2  = dsum2 + 16;        // 16
    float* fb     = (float*)((char*)d_ws + 1024);
    float* scale1 = fb;                // 32
    float* shift1 = fb + 32;           // 32
    float* scale2 = fb + 64;           // 16
    float* shift2 = fb + 80;           // 16
    float* f      = fb + 128;                       // R*256   = 4 MB
    float* h1     = f  + (size_t)RR * 256;          // R*2048  = 32 MB
    float* h2     = h1 + (size_t)RR * 2048;         // R*256   = 4 MB

    k_zero_stats <<<1, 128, 0, stream>>>(dsum1);
    k_roipool    <<<RR, 256, 0, stream>>>(img, nodes, f);
    k_conv1_stats<<<RR, 256, 0, stream>>>(f, c1w, c1b, dsum1, dsq1);
    k_bn_finalize<<<1, 64, 0, stream>>>(dsum1, dsq1, g1, bb1, scale1, shift1,
                                        32, (double)RR * 256.0);
    k_conv1_apply<<<RR, 256, 0, stream>>>(f, c1w, c1b, scale1, shift1, h1);
    k_conv2_stats<<<RR, 256, 0, stream>>>(h1, c2w, c2b, dsum2, dsq2);
    k_bn_finalize<<<1, 64, 0, stream>>>(dsum2, dsq2, g2, bb2, scale2, shift2,
                                        16, (double)RR * 64.0);
    k_conv2_apply<<<RR, 256, 0, stream>>>(h1, c2w, c2b, scale2, shift2, h2);
    k_fc_wmma    <<<RR / 16, 32, 0, stream>>>(h2, fc1w, fc1b, fc2w, fc2b, out);
}